// RegionLoss_49890340110377
// MI455X (gfx1250) — compile-verified
//
#include <hip/hip_runtime.h>
#include <math.h>

#define NB 16
#define NA 5
#define NC 20
#define NH 128
#define NW 128
#define MAXB 50
#define NBOX (NB*MAXB)            // 800
#define PLANE (NH*NW)             // 16384
#define BSTRIDE (NA*(5+NC)*PLANE) // 2048000 floats per batch
#define IGNORE_T 0.6f
#define OBJ_SCALE 5.0f

#define CONF_THREADS 256
#define TILE 8192                 // cells per block (half a plane)
#define CONF_BLOCKS (NB*NA*2)     // 160
#define CELLS_PER_THREAD (TILE/CONF_THREADS)
#define NPART (CONF_BLOCKS+1)     // 160 conf partials + 1 obj partial

typedef float v2f __attribute__((ext_vector_type(2)));
typedef float v8f __attribute__((ext_vector_type(8)));
typedef unsigned int u32x4 __attribute__((ext_vector_type(4)));
typedef int i32x4 __attribute__((ext_vector_type(4)));
typedef int i32x8 __attribute__((ext_vector_type(8)));

#if defined(__gfx1250__) && __has_builtin(__builtin_amdgcn_tensor_load_to_lds)
#define USE_TDM 1
#endif

// ---- exact f32 wave(32) reduction via the matrix pipe ------------------
// A(16x4): lane m (0-15) VGPR0=K0, lane m+16 VGPR0=K2, VGPR1=K1/K3=0.
// B(4x16)=1 => D[m][n] = v[m] + v[m+16] for all n. Sum 8 D regs per lane,
// then pair lanes l and l^16 -> full 32-lane sum in every lane.
__device__ __forceinline__ float wave_reduce_f32(float v) {
#if defined(__gfx1250__) && __has_builtin(__builtin_amdgcn_wmma_f32_16x16x4_f32)
    v2f a; a.x = v;    a.y = 0.0f;
    v2f b; b.x = 1.0f; b.y = 1.0f;
    v8f c = {0.f,0.f,0.f,0.f,0.f,0.f,0.f,0.f};
    v8f d = __builtin_amdgcn_wmma_f32_16x16x4_f32(false, a, false, b,
                                                  (short)0, c, false, false);
    float s = 0.0f;
    #pragma unroll
    for (int i = 0; i < 8; ++i) s += d[i];
    s += __shfl_xor(s, 16, 32);
    return s;
#else
    #pragma unroll
    for (int o = 16; o > 0; o >>= 1) v += __shfl_xor(v, o, 32);
    return v;
#endif
}

// ---- packed per-box meta: (gj*128+gi) | best_a<<14 | ignore5<<17 -------
__device__ __forceinline__ int box_meta(const float* __restrict__ tgt,
                                        const float* __restrict__ anc,
                                        int box /*global id*/) {
    const float* t = tgt + (size_t)box * 5;
    float gx = t[1] * (float)NW, gy = t[2] * (float)NH;
    float gw = t[3] * 1024.0f,   gh = t[4] * 1024.0f;
    int gi = (int)gx, gj = (int)gy;
    int best = 0, ign = 0;
    float bestv = -1.0f, area = gw * gh;
    #pragma unroll
    for (int a = 0; a < NA; ++a) {
        float aw = anc[2*a], ah = anc[2*a+1];
        float inter = fminf(gw, aw) * fminf(gh, ah);
        float iou = inter / (area + aw*ah - inter);
        if (iou > bestv) { bestv = iou; best = a; }
        if (iou > IGNORE_T) ign |= (1 << a);
    }
    return (gj*NW + gi) | (best << 14) | (ign << 17);
}

// ---- Kernel 1: confidence loss over all 1.31M cells --------------------
__global__ __launch_bounds__(CONF_THREADS)
void conf_kernel(const float* __restrict__ out, const float* __restrict__ tgt,
                 const float* __restrict__ anc, float* __restrict__ partials) {
#ifdef USE_TDM
    __shared__ float sconf[TILE];       // 32 KB staged by the TDM
#endif
    __shared__ int   smeta[MAXB];
    __shared__ float swave[CONF_THREADS/32];

    const int blk  = blockIdx.x;
    const int b    = blk / (NA*2);
    const int a    = (blk >> 1) % NA;
    const int half = blk & 1;
    const int tid  = threadIdx.x;
    const float* src = out + (size_t)b*BSTRIDE + (size_t)(a*25+4)*PLANE
                           + (size_t)half*TILE;

#ifdef USE_TDM
    if (tid == 0) {
        unsigned long long ga = (unsigned long long)(size_t)src;
        unsigned lds = (unsigned)(size_t)&sconf[0];
        u32x4 g0; i32x8 g1; i32x4 g2, g3; i32x8 g4;
        g0[0] = 1u;                                   // count=1, user mode
        g0[1] = lds;                                  // lds_addr
        g0[2] = (unsigned)(ga & 0xFFFFFFFFu);         // global_addr lo
        g0[3] = (unsigned)(ga >> 32) | (2u << 30);    // global_addr hi | type=2
        g1[0] = 0x00020000;                           // data_size=4B
        g1[1] = (int)((TILE & 0xFFFF) << 16);         // tensor_dim0 lo16
        g1[2] = (int)((TILE >> 16) | (1u << 16));     // tensor_dim0 hi | tensor_dim1=1
        g1[3] = (int)((unsigned)TILE << 16);          // tile_dim0 = 8192
        g1[4] = 1;                                    // tile_dim1=1, tile_dim2=0
        g1[5] = TILE;                                 // tensor_dim0_stride lo32
        g1[6] = (int)((unsigned)TILE << 16);          // stride0 hi | stride1 lo16
        g1[7] = 0;
        g2[0]=g2[1]=g2[2]=g2[3]=0; g3[0]=g3[1]=g3[2]=g3[3]=0;
        g4[0]=g4[1]=g4[2]=g4[3]=g4[4]=g4[5]=g4[6]=g4[7]=0;
        __builtin_amdgcn_tensor_load_to_lds(g0, g1, g2, g3, g4, 0);
    }
#endif
    if (tid < MAXB) smeta[tid] = box_meta(tgt, anc, b*MAXB + tid);
#ifdef USE_TDM
    if (tid == 0) __builtin_amdgcn_s_wait_tensorcnt(0);
#endif
    __syncthreads();

    float p = 0.0f;
    for (int k = 0; k < CELLS_PER_THREAD; ++k) {
        int s = tid + k*CONF_THREADS;
#ifdef USE_TDM
        float o = sconf[s];
#else
        float o = src[s];
#endif
        int planeLocal = half*TILE + s;
        bool isObj = false, isIgn = false;
        #pragma unroll 5
        for (int q = 0; q < MAXB; ++q) {
            int m = smeta[q];
            if ((m & 0x3FFF) == planeLocal) {
                if (((m >> 14) & 7) == a) isObj = true;
                if ((m >> (17 + a)) & 1)  isIgn = true;
            }
        }
        float conf = 1.0f / (1.0f + expf(-o));
        float d1 = conf - 1.0f;
        p += isObj ? (0.5f*OBJ_SCALE*d1*d1) : (isIgn ? 0.0f : 0.5f*conf*conf);
    }

    float ws = wave_reduce_f32(p);         // all 256 threads: EXEC all ones
    if ((tid & 31) == 0) swave[tid >> 5] = ws;
    __syncthreads();
    if (tid == 0) {
        float s = 0.0f;
        #pragma unroll
        for (int i = 0; i < CONF_THREADS/32; ++i) s += swave[i];
        partials[blk] = s;
    }
}

// ---- Kernel 2: coord + class loss at winner obj cells (<=800) ----------
__global__ __launch_bounds__(1024)
void obj_kernel(const float* __restrict__ out, const float* __restrict__ tgt,
                const float* __restrict__ anc, float* __restrict__ partials) {
    __shared__ int   smeta[NBOX];
    __shared__ float red[1024];
    const int tid = threadIdx.x;
    if (tid < NBOX) smeta[tid] = box_meta(tgt, anc, tid);
    __syncthreads();

    float loss = 0.0f;
    if (tid < NBOX) {
        const int b = tid / MAXB;
        const int m = smeta[tid];
        const int key = m & 0x1FFFF;       // cell + best_a
        bool winner = true;                // last (highest-index) writer wins
        for (int q = tid + 1; q < (b+1)*MAXB; ++q)
            if ((smeta[q] & 0x1FFFF) == key) { winner = false; break; }
        if (winner) {
            const float* t = tgt + (size_t)tid * 5;
            float gx = t[1]*(float)NW, gy = t[2]*(float)NH;
            float gw = t[3]*1024.0f,   gh = t[4]*1024.0f;
            int gi = (int)gx, gj = (int)gy;
            int best = (m >> 14) & 7;
            size_t base = (size_t)b*BSTRIDE + (size_t)best*25*PLANE
                        + (size_t)gj*NW + gi;
            float ox = out[base], oy = out[base + PLANE];
            float ow = out[base + 2*(size_t)PLANE], oh = out[base + 3*(size_t)PLANE];
            float x = 1.0f/(1.0f+expf(-ox)), y = 1.0f/(1.0f+expf(-oy));
            float tx = gx - floorf(gx), ty = gy - floorf(gy);
            float tw = logf(gw / anc[2*best]), th = logf(gh / anc[2*best+1]);
            float dx=x-tx, dy=y-ty, dw=ow-tw, dh=oh-th;
            loss = 0.5f*(dx*dx + dy*dy + dw*dw + dh*dh);
            // log-softmax over 20 classes
            float lg[NC]; float mx = -1e30f;
            #pragma unroll
            for (int c = 0; c < NC; ++c) {
                lg[c] = out[base + (size_t)(5+c)*PLANE];
                mx = fmaxf(mx, lg[c]);
            }
            float se = 0.0f;
            #pragma unroll
            for (int c = 0; c < NC; ++c) se += expf(lg[c] - mx);
            int lab = (int)t[0];
            loss += -(lg[lab] - mx - logf(se));   // CLASS_SCALE = 1
        }
    }
    red[tid] = loss;
    __syncthreads();
    #pragma unroll
    for (int s = 512; s > 0; s >>= 1) {
        if (tid < s) red[tid] += red[tid + s];
        __syncthreads();
    }
    if (tid == 0) partials[CONF_BLOCKS] = red[0];
}

// ---- Kernel 3: deterministic final reduction of 161 partials -----------
__global__ __launch_bounds__(32)
void final_kernel(const float* __restrict__ partials, float* __restrict__ o) {
    const int lane = threadIdx.x;
    float s = 0.0f;
    for (int i = lane; i < NPART; i += 32) s += partials[i];
    s = wave_reduce_f32(s);
    if (lane == 0) o[0] = s;
}

extern "C" void kernel_launch(void* const* d_in, const int* in_sizes, int n_in,
                              void* d_out, int out_size, void* d_ws, size_t ws_size,
                              hipStream_t stream) {
    const float* out = (const float*)d_in[0];   // (16,125,128,128)
    const float* tgt = (const float*)d_in[1];   // (16,50,5)
    const float* anc = (const float*)d_in[2];   // (5,2)
    float* partials = (float*)d_ws;             // 161 floats
    conf_kernel <<<CONF_BLOCKS, CONF_THREADS, 0, stream>>>(out, tgt, anc, partials);
    obj_kernel  <<<1, 1024, 0, stream>>>(out, tgt, anc, partials);
    final_kernel<<<1, 32,  0, stream>>>(partials, (float*)d_out);
}